// AnchorMatcher_23845658428393
// MI455X (gfx1250) — compile-verified
//
#include <hip/hip_runtime.h>
#include <math.h>

// ---------------------------------------------------------------------------
// AnchorMatcher for MI455X (gfx1250, wave32).
// Memory/VALU-bound (~8MB traffic -> ~0.35us floor at 23.3 TB/s). No matmul
// -> WMMA inapplicable. CDNA5 paths used: global_load_async_to_lds_b128 +
// s_wait_asynccnt for GT staging, ds_max_u64 LDS atomics for per-GT argmax,
// global_prefetch_b8 + TH_NT (nontemporal) hints for the streamed anchor /
// target arrays. ILP=2 anchors per thread.
// ---------------------------------------------------------------------------

typedef float f32x4 __attribute__((ext_vector_type(4)));

#define MAXM 128
#define BLK  256
#define IPT  2              // anchors per thread
#define TILE (BLK * IPT)    // anchors per block

__device__ __forceinline__ f32x4 encode_box(const f32x4 an, const f32x4 g) {
  const float eps = 1.1920928955078125e-07f;  // FLT_EPSILON (jnp.finfo(f32).eps)
  float ax = (an.x + an.z) * 0.5f;
  float ay = (an.y + an.w) * 0.5f;
  float aw = fmaxf(an.z - an.x, eps);
  float ah = fmaxf(an.w - an.y, eps);
  float gx = (g.x + g.z) * 0.5f;
  float gy = (g.y + g.w) * 0.5f;
  float gw = g.z - g.x;
  float gh = g.w - g.y;
  f32x4 r;
  r.x = (gx - ax) / aw;
  r.y = (gy - ay) / ah;
  r.z = logf(gw / aw);
  r.w = logf(gh / ah);
  return r;
}

__device__ __forceinline__ float iou_pair(const f32x4 a, float areaA,
                                          const f32x4 b, float areaB) {
  float lx = fmaxf(a.x, b.x);
  float ly = fmaxf(a.y, b.y);
  float rx = fminf(a.z, b.z);
  float ry = fminf(a.w, b.w);
  float w = fmaxf(rx - lx, 0.0f);
  float h = fmaxf(ry - ly, 0.0f);
  float inter = w * h;
  return inter / (areaA + areaB - inter);
}

__global__ void am_init_winners(unsigned long long* __restrict__ gwin, int M) {
  int t = threadIdx.x;
  if (t < M) gwin[t] = 0ULL;
}

__global__ __launch_bounds__(BLK) void am_main(
    const f32x4* __restrict__ anchors, const f32x4* __restrict__ gt,
    const int* __restrict__ labels, int N, int M,
    float* __restrict__ out_cls, f32x4* __restrict__ out_reg,
    float* __restrict__ out_pos, unsigned long long* __restrict__ gwin) {
  __shared__ f32x4 sgt[MAXM];
  __shared__ float sareab[MAXM];
  __shared__ unsigned long long swin[MAXM];

  const int tid = threadIdx.x;

  // --- Stage GT boxes (2KB) into LDS via CDNA5 async DMA path -------------
  if (tid < M) {
    // Generic pointer to LDS: low 32 bits are the LDS byte address.
    unsigned lds_off = (unsigned)(unsigned long long)&sgt[tid];
    const f32x4* src = gt + tid;
    asm volatile("global_load_async_to_lds_b128 %0, %1, off"
                 :: "v"(lds_off), "v"(src)
                 : "memory");
    swin[tid] = 0ULL;
  }
  asm volatile("s_wait_asynccnt 0x0" ::: "memory");
  __syncthreads();
  if (tid < M) {
    f32x4 b = sgt[tid];
    sareab[tid] = (b.z - b.x) * (b.w - b.y);
  }
  __syncthreads();

  // Two anchors per thread, both halves coalesced per wave.
  const int i0 = blockIdx.x * TILE + tid;
  const int i1 = i0 + BLK;
  const bool v0 = (i0 < N);
  const bool v1 = (i1 < N);

  // Streaming anchors: prefetch the tile region 4 blocks ahead (speculative, OOB-safe).
  if (v0) __builtin_prefetch((const void*)(anchors + i0 + 4 * TILE), 0, 0);

  f32x4 a0 = v0 ? __builtin_nontemporal_load(anchors + i0) : (f32x4){0.f, 0.f, 0.f, 0.f};
  f32x4 a1 = v1 ? __builtin_nontemporal_load(anchors + i1) : (f32x4){0.f, 0.f, 0.f, 0.f};
  const float areaA0 = (a0.z - a0.x) * (a0.w - a0.y);
  const float areaA1 = (a1.z - a1.x) * (a1.w - a1.y);
  // Tie-break payload: larger key == smaller anchor index (first occurrence
  // of the column max wins, matching jnp.argmax over axis 0). i0 < i1.
  const unsigned long long tie0 = (unsigned long long)(~(unsigned)i0);
  const unsigned long long tie1 = (unsigned long long)(~(unsigned)i1);

  float best0 = -1.0f, best1 = -1.0f;
  int bj0 = 0, bj1 = 0;

  for (int j = 0; j < M; ++j) {
    f32x4 b = sgt[j];
    float areaB = sareab[j];
    float iou0 = iou_pair(a0, areaA0, b, areaB);
    float iou1 = iou_pair(a1, areaA1, b, areaB);
    if (iou0 > best0) { best0 = iou0; bj0 = j; }  // '>' keeps first max (argmax)
    if (iou1 > best1) { best1 = iou1; bj1 = j; }
    // IoU >= 0 -> float bits monotone as u32. Combine both candidates, filter
    // with a plain LDS read, contend ds_max_u64 only when we might win.
    unsigned long long k0 =
        v0 ? (((unsigned long long)__float_as_uint(iou0) << 32) | tie0) : 0ULL;
    unsigned long long k1 =
        v1 ? (((unsigned long long)__float_as_uint(iou1) << 32) | tie1) : 0ULL;
    unsigned long long k = (k0 > k1) ? k0 : k1;
    if (k > swin[j]) atomicMax(&swin[j], k);
  }
  __syncthreads();

  // One global u64 atomicMax per GT per block (~50K total device-wide).
  if (tid < M) {
    unsigned long long k = swin[tid];
    if (k) atomicMax(&gwin[tid], k);
  }

  if (v0) {
    bool pos = best0 >= 0.5f;   // POS_THR
    bool neg = best0 < 0.4f;    // NEG_THR (implies !pos here)
    __builtin_nontemporal_store(pos ? 1.0f : 0.0f, out_pos + i0);
    __builtin_nontemporal_store(
        pos ? (float)labels[bj0] : (neg ? 0.0f : -1.0f), out_cls + i0);
    f32x4 r = {0.f, 0.f, 0.f, 0.f};
    if (pos) r = encode_box(a0, sgt[bj0]);
    __builtin_nontemporal_store(r, out_reg + i0);
  }
  if (v1) {
    bool pos = best1 >= 0.5f;
    bool neg = best1 < 0.4f;
    __builtin_nontemporal_store(pos ? 1.0f : 0.0f, out_pos + i1);
    __builtin_nontemporal_store(
        pos ? (float)labels[bj1] : (neg ? 0.0f : -1.0f), out_cls + i1);
    f32x4 r = {0.f, 0.f, 0.f, 0.f};
    if (pos) r = encode_box(a1, sgt[bj1]);
    __builtin_nontemporal_store(r, out_reg + i1);
  }
}

__global__ void am_finalize(
    const f32x4* __restrict__ anchors, const f32x4* __restrict__ gt,
    const int* __restrict__ labels, int N, int M,
    float* __restrict__ out_cls, f32x4* __restrict__ out_reg,
    float* __restrict__ out_pos, const unsigned long long* __restrict__ gwin) {
  int g = threadIdx.x;
  if (g >= M) return;
  unsigned long long k = gwin[g];
  int a_idx = (int)(~(unsigned)(k & 0xffffffffULL));
  if (a_idx < 0 || a_idx >= N) return;

  // Reference semantics: the forced-positive anchor still uses ITS OWN matched
  // gt (argmax over gts for that anchor), not necessarily g. Recompute (128 it).
  f32x4 a = anchors[a_idx];
  float areaA = (a.z - a.x) * (a.w - a.y);
  float best = -1.0f;
  int bj = 0;
  for (int j = 0; j < M; ++j) {
    f32x4 b = gt[j];
    float areaB = (b.z - b.x) * (b.w - b.y);
    float iou = iou_pair(a, areaA, b, areaB);
    if (iou > best) { best = iou; bj = j; }
  }
  // Multiple gts may elect the same anchor; all writes are value-identical.
  out_pos[a_idx] = 1.0f;
  out_cls[a_idx] = (float)labels[bj];
  out_reg[a_idx] = encode_box(a, gt[bj]);
}

extern "C" void kernel_launch(void* const* d_in, const int* in_sizes, int n_in,
                              void* d_out, int out_size, void* d_ws, size_t ws_size,
                              hipStream_t stream) {
  const f32x4* anchors = (const f32x4*)d_in[0];        // (N,4) f32
  const f32x4* gt      = (const f32x4*)d_in[1];        // (M,4) f32
  const int*   labels  = (const int*)d_in[2];          // (M,)  i32

  const int N = in_sizes[0] / 4;
  const int M = in_sizes[1] / 4;   // 128

  float* out      = (float*)d_out;
  float* out_cls  = out;                               // N floats
  f32x4* out_reg  = (f32x4*)(out + N);                 // 4N floats (16B aligned: N%4==0)
  float* out_pos  = out + 5ll * N;                     // N floats

  unsigned long long* gwin = (unsigned long long*)d_ws;  // 128 * u64 = 1KB

  am_init_winners<<<1, MAXM, 0, stream>>>(gwin, M);

  int blocks = (N + TILE - 1) / TILE;
  am_main<<<blocks, BLK, 0, stream>>>(anchors, gt, labels, N, M,
                                      out_cls, out_reg, out_pos, gwin);

  am_finalize<<<1, MAXM, 0, stream>>>(anchors, gt, labels, N, M,
                                      out_cls, out_reg, out_pos, gwin);
}